// TemporalAttnLayer_13838384628055
// MI455X (gfx1250) — compile-verified
//
#include <hip/hip_runtime.h>
#include <math.h>

#define DIM    100
#define N_DST  32768
#define N_EDGE 524288
#define LN_EPS 1e-5f

typedef __attribute__((ext_vector_type(2))) float v2f;
typedef __attribute__((ext_vector_type(8))) float v8f;

__device__ __forceinline__ v8f wmma_f32_k4(v2f a, v2f b, v8f c) {
  // D = A(16x4 f32) * B(4x16 f32) + C(16x16 f32)
  return __builtin_amdgcn_wmma_f32_16x16x4_f32(false, a, false, b, (short)0, c,
                                               false, false);
}

// ---------------------------------------------------------------------------
// qconst[n] = wq_b[n] + sum_k cos(time_b[k]) * wq_w[n][100+k]   (zero-time term)
__global__ void k_qconst(const float* __restrict__ wq_w,
                         const float* __restrict__ wq_b,
                         const float* __restrict__ time_b,
                         float* __restrict__ qc) {
  int n = threadIdx.x;
  if (n < DIM) {
    float acc = wq_b[n];
    for (int k = 0; k < DIM; ++k)
      acc += cosf(time_b[k]) * wq_w[n * 2 * DIM + DIM + k];
    qc[n] = acc;
  }
}

// CSR segment offsets from sorted edge_dst: offsets[d] = lower_bound(edge_dst, d)
__global__ void k_offsets(const int* __restrict__ edge_dst,
                          int* __restrict__ offsets) {
  int d = blockIdx.x * blockDim.x + threadIdx.x;
  if (d > N_DST) return;
  int lo = 0, hi = N_EDGE;
  while (lo < hi) {
    int mid = (lo + hi) >> 1;
    if (edge_dst[mid] < d) lo = mid + 1; else hi = mid;
  }
  offsets[d] = lo;
}

// ---------------------------------------------------------------------------
// Q[32768 x 100] = dst_h @ wq_w[:, :100]^T + qconst
// 8 waves: 2 row-tiles x 4 col-tiles (32 rows x 64 cols), grid.y covers N.
__global__ __launch_bounds__(256) void k_gemm_q(const float* __restrict__ A,
                                                const float* __restrict__ W,
                                                const float* __restrict__ qc,
                                                float* __restrict__ Q) {
  __shared__ float As[32 * 104];
  const int tid = threadIdx.x;
  const int rbase = blockIdx.x * 32;
  for (int i = tid; i < 32 * DIM; i += 256)
    As[(i / DIM) * 104 + (i % DIM)] = A[rbase * DIM + i];
  __syncthreads();

  const int wave = tid >> 5, lane = tid & 31;
  const int row0 = (wave >> 2) * 16;
  const int col0 = blockIdx.y * 64 + (wave & 3) * 16;
  const int kh = (lane >> 4) << 1;      // K sub-offset 0 or 2
  const int ml = lane & 15;
  int nb = col0 + ml; if (nb > DIM - 1) nb = DIM - 1;   // clamp OOB B rows
  const float* Arow = &As[(row0 + ml) * 104 + kh];
  const float* Wrow = &W[nb * 200 + kh];                // wq_w row stride 200
  v8f c = {};
  for (int kb = 0; kb < DIM; kb += 4)
    c = wmma_f32_k4(*(const v2f*)(Arow + kb), *(const v2f*)(Wrow + kb), c);

  if (col0 + ml < DIM) {
    float bias = qc[col0 + ml];
#pragma unroll
    for (int i = 0; i < 8; ++i) {
      int r = rbase + row0 + ((lane < 16) ? i : i + 8);
      Q[r * DIM + col0 + ml] = c[i] + bias;
    }
  }
}

// ---------------------------------------------------------------------------
// Fused KV projection + attention logits. 13 waves; 16 edges x 208 cols.
// A panel = [src_h | efeat | cos(t*w+b)] staged in LDS; Z tile kept in LDS;
// emits attn[e][2] (leaky-relu(Q.K)) and V[e][100] only (Z never hits HBM).
__global__ __launch_bounds__(416) void k_gemm_z(
    const float* __restrict__ src_h, const float* __restrict__ efeat,
    const float* __restrict__ tdel, const float* __restrict__ time_w,
    const float* __restrict__ time_b, const float* __restrict__ wkv_w,
    const float* __restrict__ wkv_b, const int* __restrict__ edge_dst,
    const float* __restrict__ Q, float* __restrict__ Vout,
    float* __restrict__ attn) {
  __shared__ float As[16 * 304];
  __shared__ float Zs[16 * 208];
  const int tid = threadIdx.x;
  const int rbase = blockIdx.x * 16;

  for (int i = tid; i < 16 * DIM; i += 416) {
    int r = i / DIM, cc = i % DIM;
    As[r * 304 + cc] = src_h[rbase * DIM + i];
    As[r * 304 + DIM + cc] = efeat[rbase * DIM + i];
    As[r * 304 + 2 * DIM + cc] = cosf(tdel[rbase + r] * time_w[cc] + time_b[cc]);
  }
  __syncthreads();

  const int wave = tid >> 5, lane = tid & 31;
  const int col0 = wave * 16;                 // 0..192 (13 tiles cover N=200)
  const int kh = (lane >> 4) << 1;
  const int ml = lane & 15;
  int nb = col0 + ml; if (nb > 199) nb = 199;
  const float* Arow = &As[ml * 304 + kh];
  const float* Wrow = &wkv_w[nb * 300 + kh];
  v8f c = {};
  for (int kb = 0; kb < 300; kb += 4)
    c = wmma_f32_k4(*(const v2f*)(Arow + kb), *(const v2f*)(Wrow + kb), c);

  if (col0 + ml < 200) {
    float bias = wkv_b[col0 + ml];
#pragma unroll
    for (int i = 0; i < 8; ++i) {
      int r = (lane < 16) ? i : i + 8;
      Zs[r * 208 + col0 + ml] = c[i] + bias;
    }
  }
  __syncthreads();

  // attn[e,h] = leaky_relu( dot(Q[dst[e]][h*50:+50], K[e][h*50:+50]), 0.2 )
  if (tid < 32) {
    int r = tid >> 1, h = tid & 1;
    int e = rbase + r;
    const float* qrow = &Q[edge_dst[e] * DIM + h * 50];
    const float* krow = &Zs[r * 208 + h * 50];
    float s = 0.f;
    for (int j = 0; j < 50; ++j) s += qrow[j] * krow[j];
    attn[e * 2 + h] = (s > 0.f) ? s : 0.2f * s;
  }
  // spill V (cols 100..199), coalesced
  for (int i = tid; i < 16 * DIM; i += 416) {
    int r = i / DIM, cc = i % DIM;
    Vout[rbase * DIM + i] = Zs[r * 208 + DIM + cc];
  }
}

// ---------------------------------------------------------------------------
// One wave32 per destination node: segment softmax + weighted V reduce.
__global__ __launch_bounds__(256) void k_agg(const float* __restrict__ attn,
                                             const float* __restrict__ V,
                                             const int* __restrict__ offsets,
                                             float* __restrict__ agg) {
  const int lane = threadIdx.x & 31;
  const int d = blockIdx.x * 8 + (threadIdx.x >> 5);
  const int lo = offsets[d], hi = offsets[d + 1];

  float m0 = -INFINITY, m1 = -INFINITY;
  for (int e = lo + lane; e < hi; e += 32) {
    v2f a = *(const v2f*)(attn + 2 * e);
    m0 = fmaxf(m0, a.x); m1 = fmaxf(m1, a.y);
  }
  for (int o = 16; o > 0; o >>= 1) {
    m0 = fmaxf(m0, __shfl_xor(m0, o, 32));
    m1 = fmaxf(m1, __shfl_xor(m1, o, 32));
  }
  float s0 = 0.f, s1 = 0.f;
  for (int e = lo + lane; e < hi; e += 32) {
    v2f a = *(const v2f*)(attn + 2 * e);
    s0 += expf(a.x - m0); s1 += expf(a.y - m1);
  }
  for (int o = 16; o > 0; o >>= 1) {
    s0 += __shfl_xor(s0, o, 32);
    s1 += __shfl_xor(s1, o, 32);
  }
  const float i0 = 1.f / s0, i1 = 1.f / s1;

  float a0 = 0.f, a1 = 0.f, a2 = 0.f, a3 = 0.f;
  for (int e = lo; e < hi; ++e) {
    v2f a = *(const v2f*)(attn + 2 * e);           // uniform broadcast load
    float w0 = expf(a.x - m0) * i0;
    float w1 = expf(a.y - m1) * i1;
    const float* vr = V + e * DIM;
    a0 += w0 * vr[lane];                           // j 0..31   head0
    a1 += ((32 + lane < 50) ? w0 : w1) * vr[32 + lane];  // head boundary @50
    a2 += w1 * vr[64 + lane];                      // j 64..95  head1
    if (96 + lane < DIM) a3 += w1 * vr[96 + lane];
  }
  float* orow = agg + d * DIM;
  orow[lane] = a0; orow[32 + lane] = a1; orow[64 + lane] = a2;
  if (96 + lane < DIM) orow[96 + lane] = a3;
}

// ---------------------------------------------------------------------------
// out = LN(relu(concat(agg, dst_h) @ wout_w^T + wout_b)) * g + b
// 8 waves = 1 row-tile (16 rows) x 8 col-tiles; LN fused via LDS row stats.
__global__ __launch_bounds__(256) void k_final(
    const float* __restrict__ agg, const float* __restrict__ dst_h,
    const float* __restrict__ wout_w, const float* __restrict__ wout_b,
    const float* __restrict__ ln_g, const float* __restrict__ ln_b,
    float* __restrict__ out) {
  __shared__ float As[16 * 208];
  __shared__ float Os[16 * 128];
  __shared__ float mu_s[16], rs_s[16];
  const int tid = threadIdx.x;
  const int rbase = blockIdx.x * 16;

  for (int i = tid; i < 16 * DIM; i += 256) {
    int r = i / DIM, cc = i % DIM;
    As[r * 208 + cc] = agg[rbase * DIM + i];
    As[r * 208 + DIM + cc] = dst_h[rbase * DIM + i];
  }
  __syncthreads();

  const int wave = tid >> 5, lane = tid & 31;
  const int col0 = wave * 16;                 // 0..112
  const int kh = (lane >> 4) << 1;
  const int ml = lane & 15;
  int nb = col0 + ml; if (nb > DIM - 1) nb = DIM - 1;
  const float* Arow = &As[ml * 208 + kh];
  const float* Wrow = &wout_w[nb * 200 + kh];
  v8f c = {};
  for (int kb = 0; kb < 200; kb += 4)
    c = wmma_f32_k4(*(const v2f*)(Arow + kb), *(const v2f*)(Wrow + kb), c);

  if (col0 + ml < DIM) {
    float bias = wout_b[col0 + ml];
#pragma unroll
    for (int i = 0; i < 8; ++i) {
      int r = (lane < 16) ? i : i + 8;
      Os[r * 128 + col0 + ml] = fmaxf(c[i] + bias, 0.f);
    }
  }
  __syncthreads();

  if (tid < 16) {
    float s = 0.f, ss = 0.f;
    for (int j = 0; j < DIM; ++j) { float v = Os[tid * 128 + j]; s += v; ss += v * v; }
    float mu = s * (1.f / DIM);
    float var = ss * (1.f / DIM) - mu * mu;
    mu_s[tid] = mu;
    rs_s[tid] = rsqrtf(var + LN_EPS);
  }
  __syncthreads();

  for (int i = tid; i < 16 * DIM; i += 256) {
    int r = i / DIM, cc = i % DIM;
    out[rbase * DIM + i] = (Os[r * 128 + cc] - mu_s[r]) * rs_s[r] * ln_g[cc] + ln_b[cc];
  }
}

// ---------------------------------------------------------------------------
static inline size_t alignup(size_t x) { return (x + 255) & ~(size_t)255; }

extern "C" void kernel_launch(void* const* d_in, const int* in_sizes, int n_in,
                              void* d_out, int out_size, void* d_ws, size_t ws_size,
                              hipStream_t stream) {
  (void)in_sizes; (void)n_in; (void)out_size; (void)ws_size;
  const float* dst_h  = (const float*)d_in[0];
  const float* src_h  = (const float*)d_in[1];
  const float* efeat  = (const float*)d_in[2];
  const float* tdel   = (const float*)d_in[3];
  const int*   edst   = (const int*)  d_in[4];
  const float* time_w = (const float*)d_in[5];
  const float* time_b = (const float*)d_in[6];
  const float* wq_w   = (const float*)d_in[7];
  const float* wq_b   = (const float*)d_in[8];
  const float* wkv_w  = (const float*)d_in[9];
  const float* wkv_b  = (const float*)d_in[10];
  const float* wout_w = (const float*)d_in[11];
  const float* wout_b = (const float*)d_in[12];
  const float* ln_g   = (const float*)d_in[13];
  const float* ln_b   = (const float*)d_in[14];
  float* out = (float*)d_out;

  char* ws = (char*)d_ws;
  size_t off = 0;
  float* qc   = (float*)(ws + off); off = alignup(off + DIM * sizeof(float));
  int*   offs = (int*)  (ws + off); off = alignup(off + (size_t)(N_DST + 1) * sizeof(int));
  float* Q    = (float*)(ws + off); off = alignup(off + (size_t)N_DST * DIM * sizeof(float));
  float* attn = (float*)(ws + off); off = alignup(off + (size_t)N_EDGE * 2 * sizeof(float));
  float* agg  = (float*)(ws + off); off = alignup(off + (size_t)N_DST * DIM * sizeof(float));
  float* V    = (float*)(ws + off); off = alignup(off + (size_t)N_EDGE * DIM * sizeof(float));

  k_qconst <<<1, 128, 0, stream>>>(wq_w, wq_b, time_b, qc);
  k_offsets<<<(N_DST + 256) / 256, 256, 0, stream>>>(edst, offs);
  k_gemm_q <<<dim3(N_DST / 32, 2), 256, 0, stream>>>(dst_h, wq_w, qc, Q);
  k_gemm_z <<<N_EDGE / 16, 416, 0, stream>>>(src_h, efeat, tdel, time_w, time_b,
                                             wkv_w, wkv_b, edst, Q, V, attn);
  k_agg    <<<N_DST / 8, 256, 0, stream>>>(attn, V, offs, agg);
  k_final  <<<N_DST / 16, 256, 0, stream>>>(agg, dst_h, wout_w, wout_b, ln_g, ln_b, out);
}